// HierarchicalGNN_5076651344320
// MI455X (gfx1250) — compile-verified
//
#include <hip/hip_runtime.h>

typedef __attribute__((ext_vector_type(2))) float v2f;
typedef __attribute__((ext_vector_type(8))) float v8f;

#define EPS 1e-5f

// ---------------------------------------------------------------------------
// Generic fp32 WMMA GEMM: C[M x NCOL] = A[M x K] * B[K x NCOL] (+bias)(relu)
// A may be split as concat(A1[M x 256], A2[M x 256]) when A2 != nullptr (K=512).
// Block = 256 threads (8 waves). Block covers 16 rows x NCOL cols.
// A panel staged in LDS with padded stride (K+4) to avoid 64-bank conflicts.
// Each wave computes 16x16 tiles with V_WMMA_F32_16X16X4_F32, striding col tiles.
// ---------------------------------------------------------------------------
__global__ __launch_bounds__(256)
void wmma_gemm(const float* __restrict__ A1, const float* __restrict__ A2,
               const float* __restrict__ B, const float* __restrict__ bias,
               float* __restrict__ C, int M, int K, int NCOL, int relu) {
  __shared__ float lds[16 * 516];  // supports K up to 512 (+4 pad)
  const int KP = K + 4;
  const int row0 = blockIdx.x * 16;
  const int tid = threadIdx.x;

  // Stage A panel (coalesced along K)
  const int total = 16 * K;
  for (int idx = tid; idx < total; idx += 256) {
    int r = idx / K;
    int k = idx - r * K;
    int grow = row0 + r;
    float v = 0.0f;
    if (grow < M) {
      if (A2 != nullptr && k >= 256) {
        v = A2[(size_t)grow * 256 + (k - 256)];
      } else {
        int lda = (A2 != nullptr) ? 256 : K;
        v = A1[(size_t)grow * lda + k];
      }
    }
    lds[r * KP + k] = v;
  }
  __syncthreads();

  const int wid  = tid >> 5;
  const int lane = tid & 31;
  const int arow = lane & 15;   // A row (M) for this lane
  const int khi  = lane >> 4;   // 0: K pair {0,1}; 1: K pair {2,3}
  const int col  = lane & 15;   // B/C column within tile
  const int ntiles = NCOL >> 4;

  for (int ct = wid; ct < ntiles; ct += 8) {
    const int col0 = ct << 4;
    v8f acc = {};
    for (int kc = 0; kc < K; kc += 4) {
      v2f a, b;
      // 32-bit A-matrix 16x4 layout: lanes 0-15 hold K={0,1}, lanes 16-31 K={2,3}
      a.x = lds[arow * KP + kc + 2 * khi];
      a.y = lds[arow * KP + kc + 2 * khi + 1];
      // B 4x16: row striped across lanes within a VGPR, K pairs mirrored
      b.x = B[(size_t)(kc + 2 * khi) * NCOL + col0 + col];
      b.y = B[(size_t)(kc + 2 * khi + 1) * NCOL + col0 + col];
      acc = __builtin_amdgcn_wmma_f32_16x16x4_f32(false, a, false, b,
                                                  (short)0, acc, false, false);
    }
    // C/D layout: VGPR r -> row (r + 8*khi), lanes%16 -> column
#pragma unroll
    for (int r = 0; r < 8; ++r) {
      int orow = row0 + r + 8 * khi;
      if (orow < M) {
        int ocol = col0 + col;
        float v = acc[r];
        if (bias) v += bias[ocol];
        if (relu) v = fmaxf(v, 0.0f);
        C[(size_t)orow * NCOL + ocol] = v;
      }
    }
  }
}

// ---------------------------------------------------------------------------
// Per-node per-head attention scores: a_src[n,h] = sum_c hW[n,h*64+c]*att_src[h,c]
// Warp per node; lane owns 8 consecutive channels; head = lane/8.
// ---------------------------------------------------------------------------
__global__ __launch_bounds__(256)
void att_scores(const float* __restrict__ hW, const float* __restrict__ avS,
                const float* __restrict__ avD, float* __restrict__ asrc,
                float* __restrict__ adst, int n) {
  int warp = (blockIdx.x * blockDim.x + threadIdx.x) >> 5;
  int lane = threadIdx.x & 31;
  if (warp >= n) return;
  const float* row = hW + (size_t)warp * 256 + lane * 8;
  float s = 0.0f, d = 0.0f;
#pragma unroll
  for (int i = 0; i < 8; ++i) {
    float v = row[i];
    s += v * avS[lane * 8 + i];
    d += v * avD[lane * 8 + i];
  }
  // reduce within 8-lane head groups
#pragma unroll
  for (int off = 1; off <= 4; off <<= 1) {
    s += __shfl_xor(s, off, 32);
    d += __shfl_xor(d, off, 32);
  }
  if ((lane & 7) == 0) {
    int head = lane >> 3;
    asrc[warp * 4 + head] = s;
    adst[warp * 4 + head] = d;
  }
}

// Monotone float->uint key for atomicMax over signed floats
__device__ __forceinline__ unsigned orderKey(float f) {
  unsigned u = __float_as_uint(f);
  return (u & 0x80000000u) ? ~u : (u | 0x80000000u);
}
__device__ __forceinline__ float decodeKey(unsigned k) {
  unsigned u = (k & 0x80000000u) ? (k & 0x7fffffffu) : ~k;
  return __uint_as_float(u);
}

// Edge pass 1: e = leaky_relu(a_src[src]+a_dst[dst]); segment max via atomicMax.
// t in [0,E): real edge; t in [E,E+n): self loop (src=dst=t-E).
__global__ __launch_bounds__(256)
void edge_scores(const int* __restrict__ ei, const float* __restrict__ asrc,
                 const float* __restrict__ adst, float* __restrict__ esc,
                 unsigned* __restrict__ mkey, int E, int n) {
  int t = blockIdx.x * blockDim.x + threadIdx.x;
  int T = E + n;
  if (t >= T) return;
  int src, dst;
  if (t < E) { src = ei[t]; dst = ei[E + t]; } else { src = dst = t - E; }
#pragma unroll
  for (int h = 0; h < 4; ++h) {
    float v = asrc[src * 4 + h] + adst[dst * 4 + h];
    v = v > 0.0f ? v : 0.2f * v;
    esc[(size_t)t * 4 + h] = v;
    atomicMax(&mkey[dst * 4 + h], orderKey(v));
  }
}

// Edge pass 2: ex = exp(e - m[dst]); denom[dst] += ex (stores ex over e in place)
__global__ __launch_bounds__(256)
void edge_exp(const int* __restrict__ ei, float* __restrict__ esc,
              const unsigned* __restrict__ mkey, float* __restrict__ denom,
              int E, int n) {
  int t = blockIdx.x * blockDim.x + threadIdx.x;
  int T = E + n;
  if (t >= T) return;
  int dst = (t < E) ? ei[E + t] : (t - E);
#pragma unroll
  for (int h = 0; h < 4; ++h) {
    float m = decodeKey(mkey[dst * 4 + h]);
    float ex = __expf(esc[(size_t)t * 4 + h] - m);
    esc[(size_t)t * 4 + h] = ex;
    atomicAdd(&denom[dst * 4 + h], ex);
  }
}

// Edge pass 3: accum[dst, :] += alpha * hW[src, :] ; warp per edge, 8 ch/lane
__global__ __launch_bounds__(256)
void edge_scatter(const int* __restrict__ ei, const float* __restrict__ hW,
                  const float* __restrict__ esc, const float* __restrict__ denom,
                  float* __restrict__ accum, int E, int n) {
  int warp = (blockIdx.x * blockDim.x + threadIdx.x) >> 5;
  int lane = threadIdx.x & 31;
  int T = E + n;
  if (warp >= T) return;
  int src, dst;
  if (warp < E) { src = ei[warp]; dst = ei[E + warp]; } else { src = dst = warp - E; }
  int head = lane >> 3;
  float alpha = esc[(size_t)warp * 4 + head] / denom[dst * 4 + head];
  const float* row = hW + (size_t)src * 256 + lane * 8;
  float* orow = accum + (size_t)dst * 256 + lane * 8;
#pragma unroll
  for (int i = 0; i < 8; ++i) atomicAdd(&orow[i], row[i] * alpha);
}

// GAT epilogue (in place): x = relu(LN(accum + bias; g, b)); warp per node
__global__ __launch_bounds__(256)
void gat_finalize(float* __restrict__ x, const float* __restrict__ bias,
                  const float* __restrict__ g, const float* __restrict__ b, int n) {
  int warp = (blockIdx.x * blockDim.x + threadIdx.x) >> 5;
  int lane = threadIdx.x & 31;
  if (warp >= n) return;
  float* row = x + (size_t)warp * 256 + lane * 8;
  float v[8];
  float s = 0.0f;
#pragma unroll
  for (int i = 0; i < 8; ++i) { v[i] = row[i] + bias[lane * 8 + i]; s += v[i]; }
#pragma unroll
  for (int off = 16; off >= 1; off >>= 1) s += __shfl_xor(s, off, 32);
  float mean = s * (1.0f / 256.0f);
  float q = 0.0f;
#pragma unroll
  for (int i = 0; i < 8; ++i) { float d = v[i] - mean; q += d * d; }
#pragma unroll
  for (int off = 16; off >= 1; off >>= 1) q += __shfl_xor(q, off, 32);
  float inv = rsqrtf(q * (1.0f / 256.0f) + EPS);
#pragma unroll
  for (int i = 0; i < 8; ++i) {
    float y = (v[i] - mean) * inv * g[lane * 8 + i] + b[lane * 8 + i];
    row[i] = fmaxf(y, 0.0f);
  }
}

// Final: out = L2norm(LN(raw + bf; g, b)); warp per node, 4 ch/lane (128 cols)
__global__ __launch_bounds__(256)
void final_norm(const float* __restrict__ raw, const float* __restrict__ bf,
                const float* __restrict__ g, const float* __restrict__ b,
                float* __restrict__ out, int n) {
  int warp = (blockIdx.x * blockDim.x + threadIdx.x) >> 5;
  int lane = threadIdx.x & 31;
  if (warp >= n) return;
  const float* row = raw + (size_t)warp * 128 + lane * 4;
  float v[4];
  float s = 0.0f;
#pragma unroll
  for (int i = 0; i < 4; ++i) { v[i] = row[i] + bf[lane * 4 + i]; s += v[i]; }
#pragma unroll
  for (int off = 16; off >= 1; off >>= 1) s += __shfl_xor(s, off, 32);
  float mean = s * (1.0f / 128.0f);
  float q = 0.0f;
#pragma unroll
  for (int i = 0; i < 4; ++i) { float d = v[i] - mean; q += d * d; }
#pragma unroll
  for (int off = 16; off >= 1; off >>= 1) q += __shfl_xor(q, off, 32);
  float inv = rsqrtf(q * (1.0f / 128.0f) + EPS);
  float y[4];
  float sq = 0.0f;
#pragma unroll
  for (int i = 0; i < 4; ++i) {
    y[i] = (v[i] - mean) * inv * g[lane * 4 + i] + b[lane * 4 + i];
    sq += y[i] * y[i];
  }
#pragma unroll
  for (int off = 16; off >= 1; off >>= 1) sq += __shfl_xor(sq, off, 32);
  float nrm = fmaxf(sqrtf(sq), 1e-12f);
  float* orow = out + (size_t)warp * 128 + lane * 4;
#pragma unroll
  for (int i = 0; i < 4; ++i) orow[i] = y[i] / nrm;
}

// ---------------------------------------------------------------------------
extern "C" void kernel_launch(void* const* d_in, const int* in_sizes, int n_in,
                              void* d_out, int out_size, void* d_ws, size_t ws_size,
                              hipStream_t stream) {
  (void)n_in; (void)out_size; (void)ws_size;
  const float* x       = (const float*)d_in[0];
  const int*   ei_bu   = (const int*)d_in[1];
  const int*   ei_td   = (const int*)d_in[2];
  const float* Wp      = (const float*)d_in[3];
  const float* bp      = (const float*)d_in[4];
  const float* W_bu    = (const float*)d_in[5];
  const float* as_bu   = (const float*)d_in[6];
  const float* ad_bu   = (const float*)d_in[7];
  const float* bias_bu = (const float*)d_in[8];
  const float* W_td    = (const float*)d_in[9];
  const float* as_td   = (const float*)d_in[10];
  const float* ad_td   = (const float*)d_in[11];
  const float* bias_td = (const float*)d_in[12];
  const float* Wf      = (const float*)d_in[13];
  const float* bf      = (const float*)d_in[14];
  const float* g_bu    = (const float*)d_in[15];
  const float* b_bu    = (const float*)d_in[16];
  const float* g_td    = (const float*)d_in[17];
  const float* b_td    = (const float*)d_in[18];
  const float* g_out   = (const float*)d_in[19];
  const float* b_out   = (const float*)d_in[20];

  const int N = in_sizes[0] / 256;
  const int E = in_sizes[1] / 2;
  const int T = E + N;

  // Workspace layout (floats)
  const size_t NH = (size_t)N * 256;
  float* ws   = (float*)d_ws;
  float* h    = ws;               // N*256
  float* hW   = ws + NH;          // N*256 (reused as final-GEMM scratch)
  float* xbu  = ws + 2 * NH;      // N*256 accumulator -> x_bu
  float* xtd  = ws + 3 * NH;      // N*256 accumulator -> x_td
  float* sm   = ws + 4 * NH;
  float*    asrc  = sm;                              // N*4
  float*    adst  = asrc + (size_t)N * 4;            // N*4
  unsigned* mkey  = (unsigned*)(adst + (size_t)N * 4); // N*4
  float*    denom = (float*)(mkey + (size_t)N * 4);  // N*4
  float*    esc   = denom + (size_t)N * 4;           // (E+N)*4
  float* out2raw = hW;                               // N*128

  const int gGemm = (N + 15) / 16;
  const int gWarp = (N + 7) / 8;       // warp-per-node kernels
  const int gEdge = (T + 255) / 256;   // thread-per-edge kernels
  const int gScat = (T + 7) / 8;       // warp-per-edge kernel

  // 1) h = relu(x @ Wp + bp)
  wmma_gemm<<<gGemm, 256, 0, stream>>>(x, nullptr, Wp, bp, h, N, 256, 256, 1);

  // 2) two GAT directions, sequentially (shared hW / score buffers)
  const float* Wd[2]    = {W_bu, W_td};
  const float* asd[2]   = {as_bu, as_td};
  const float* add_[2]  = {ad_bu, ad_td};
  const float* biasd[2] = {bias_bu, bias_td};
  const float* gd[2]    = {g_bu, g_td};
  const float* bd[2]    = {b_bu, b_td};
  const int*   eid[2]   = {ei_bu, ei_td};
  float*       xout[2]  = {xbu, xtd};

  for (int dir = 0; dir < 2; ++dir) {
    wmma_gemm<<<gGemm, 256, 0, stream>>>(h, nullptr, Wd[dir], nullptr, hW,
                                         N, 256, 256, 0);
    att_scores<<<gWarp, 256, 0, stream>>>(hW, asd[dir], add_[dir], asrc, adst, N);
    hipMemsetAsync(mkey,  0, (size_t)N * 4 * sizeof(unsigned), stream);
    hipMemsetAsync(denom, 0, (size_t)N * 4 * sizeof(float), stream);
    hipMemsetAsync(xout[dir], 0, NH * sizeof(float), stream);
    edge_scores<<<gEdge, 256, 0, stream>>>(eid[dir], asrc, adst, esc, mkey, E, N);
    edge_exp<<<gEdge, 256, 0, stream>>>(eid[dir], esc, mkey, denom, E, N);
    edge_scatter<<<gScat, 256, 0, stream>>>(eid[dir], hW, esc, denom, xout[dir], E, N);
    gat_finalize<<<gWarp, 256, 0, stream>>>(xout[dir], biasd[dir], gd[dir], bd[dir], N);
  }

  // 3) out2raw = concat(x_bu, x_td) @ Wf   (K = 512, NCOL = 128)
  wmma_gemm<<<gGemm, 256, 0, stream>>>(xbu, xtd, Wf, nullptr, out2raw,
                                       N, 512, 128, 0);

  // 4) bias + LN + L2 normalize -> d_out
  final_norm<<<gWarp, 256, 0, stream>>>(out2raw, bf, g_out, b_out,
                                        (float*)d_out, N);
}